// UKF_Aliter_43387759624459
// MI455X (gfx1250) — compile-verified
//
#include <hip/hip_runtime.h>
#include <math.h>

// ---------------------------------------------------------------------------
// UKF scan for MI455X (gfx1250). One workgroup (8 wave32) per trajectory.
// All per-step state in LDS (~175KB of the 320KB/WGP). Dense kernels use
// V_WMMA_F32_16X16X4_F32 (full f32 precision on the matrix unit).
// ---------------------------------------------------------------------------

typedef __attribute__((ext_vector_type(2))) float v2f;
typedef __attribute__((ext_vector_type(8))) float v8f;

#define NS     64      // states
#define NOBS   32      // observations
#define NSIG   129     // 2n+1 sigma points
#define NSP    144     // sigma rows padded to multiple of 16
#define TSTEPS 250
#define NTRAJ  128
#define DT_    0.02f

// Merwe weights: n=64, lam = 1*(64-1)-64 = -1, n+lam = 63
// Wm0 = -1/63, Wc0 = -1/63 + (1 - 1 + 2) = 2 - 1/63, c = 0.5/63
__device__ __forceinline__ float wc_of(int k) {
  if (k == 0) return 2.0f - (1.0f / 63.0f);
  return (k < NSIG) ? (0.5f / 63.0f) : 0.0f;   // zero weight on padded rows
}

// --- C[M][N] = A[M][K] * B[K][N]  (row-major LDS, f32 WMMA 16x16x4) --------
__device__ __forceinline__ void gemm_ab(const float* A, int lda,
                                        const float* B, int ldb,
                                        float* C, int ldc,
                                        int M, int N, int K) {
  const int lane = threadIdx.x & 31;
  const int wave = threadIdx.x >> 5;
  const int nw   = blockDim.x >> 5;
  const int h    = lane >> 4;       // lane half: K offset 0/2 for A, rows 0-1/2-3 for B
  const int lc   = lane & 15;
  const int nt   = N >> 4;
  const int ntiles = (M >> 4) * nt;
  for (int tile = wave; tile < ntiles; tile += nw) {
    const int m0 = (tile / nt) << 4;
    const int n0 = (tile % nt) << 4;
    v8f acc = {};
    for (int k0 = 0; k0 < K; k0 += 4) {
      v2f a, b;
      const float* ap = A + (m0 + lc) * lda + (k0 + 2 * h);
      a.x = ap[0];
      a.y = ap[1];
      b.x = B[(k0 + 2 * h)     * ldb + (n0 + lc)];
      b.y = B[(k0 + 2 * h + 1) * ldb + (n0 + lc)];
      acc = __builtin_amdgcn_wmma_f32_16x16x4_f32(false, a, false, b,
                                                  (short)0, acc, false, false);
    }
#pragma unroll
    for (int vi = 0; vi < 8; ++vi)
      C[(m0 + vi + 8 * h) * ldc + (n0 + lc)] = acc[vi];
  }
}

// --- C[M][N] = sum_k wc(k) * A[k][m] * B[k][n]   (A,B are K x {M,N}) -------
// Fuses the Merwe Wc weighting and the zero padding (k>=129) into the A frag.
__device__ __forceinline__ void gemm_atb_w(const float* A, int lda,
                                           const float* B, int ldb,
                                           float* C, int ldc,
                                           int M, int N, int K) {
  const int lane = threadIdx.x & 31;
  const int wave = threadIdx.x >> 5;
  const int nw   = blockDim.x >> 5;
  const int h    = lane >> 4;
  const int lc   = lane & 15;
  const int nt   = N >> 4;
  const int ntiles = (M >> 4) * nt;
  for (int tile = wave; tile < ntiles; tile += nw) {
    const int m0 = (tile / nt) << 4;
    const int n0 = (tile % nt) << 4;
    v8f acc = {};
    for (int k0 = 0; k0 < K; k0 += 4) {
      const int ka = k0 + 2 * h;
      v2f a, b;
      a.x = wc_of(ka)     * A[ka       * lda + (m0 + lc)];
      a.y = wc_of(ka + 1) * A[(ka + 1) * lda + (m0 + lc)];
      b.x = B[ka       * ldb + (n0 + lc)];
      b.y = B[(ka + 1) * ldb + (n0 + lc)];
      acc = __builtin_amdgcn_wmma_f32_16x16x4_f32(false, a, false, b,
                                                  (short)0, acc, false, false);
    }
#pragma unroll
    for (int vi = 0; vi < 8; ++vi)
      C[(m0 + vi + 8 * h) * ldc + (n0 + lc)] = acc[vi];
  }
}

// --- C[M][N] = A[M][K] * B[N][K]^T -----------------------------------------
__device__ __forceinline__ void gemm_abt(const float* A, int lda,
                                         const float* B, int ldb,
                                         float* C, int ldc,
                                         int M, int N, int K) {
  const int lane = threadIdx.x & 31;
  const int wave = threadIdx.x >> 5;
  const int nw   = blockDim.x >> 5;
  const int h    = lane >> 4;
  const int lc   = lane & 15;
  const int nt   = N >> 4;
  const int ntiles = (M >> 4) * nt;
  for (int tile = wave; tile < ntiles; tile += nw) {
    const int m0 = (tile / nt) << 4;
    const int n0 = (tile % nt) << 4;
    v8f acc = {};
    for (int k0 = 0; k0 < K; k0 += 4) {
      const int ka = k0 + 2 * h;
      v2f a, b;
      const float* ap = A + (m0 + lc) * lda + ka;
      a.x = ap[0];
      a.y = ap[1];
      const float* bp = B + (n0 + lc) * ldb + ka;
      b.x = bp[0];
      b.y = bp[1];
      acc = __builtin_amdgcn_wmma_f32_16x16x4_f32(false, a, false, b,
                                                  (short)0, acc, false, false);
    }
#pragma unroll
    for (int vi = 0; vi < 8; ++vi)
      C[(m0 + vi + 8 * h) * ldc + (n0 + lc)] = acc[vi];
  }
}

// ---------------------------------------------------------------------------
extern "C" __global__ __launch_bounds__(256)
void ukf_traj_kernel(const float* __restrict__ X, const float* __restrict__ Y,
                     const float* __restrict__ F, const float* __restrict__ H,
                     const float* __restrict__ Q, const float* __restrict__ R,
                     float* __restrict__ out_x, float* __restrict__ out_P,
                     float* __restrict__ ws_err) {
  // Constant operands (transposed so GEMMs are plain row-major A*B)
  __shared__ float sFt[NS][NS + 1];        // F^T
  __shared__ float sHt[NS][NOBS + 1];      // H^T
  // Filter state
  __shared__ float sP[NS][NS + 1];
  __shared__ float sW[NS][NS + 1];         // chol(63*(P+eps)) then P_pred
  __shared__ float sig[NSP][NS + 1];       // sigma pts; later reused for K Pz K^T
  __shared__ float sfm[NSP][NS + 1];       // propagated sigma / dxs
  __shared__ float szs[NSP][NOBS + 1];     // zs / dzs; rows 0..63 reused for K*Pz
  __shared__ float sPz[NOBS][NOBS + 1];
  __shared__ float sLz[NOBS][NOBS + 1];
  __shared__ float sPxz[NS][NOBS + 1];
  __shared__ float sK[NS][NOBS + 1];
  __shared__ float sx[NS], sxp[NS], szp[NOBS], sy[NOBS], sinn[NOBS], serr[NS];

  const int tid  = threadIdx.x;
  const int nthr = blockDim.x;
  const int traj = blockIdx.x;

  for (int i = tid; i < NS * NS; i += nthr) {
    int r = i / NS, c = i % NS;
    sFt[c][r] = F[i];                             // F^T
    sP[r][c]  = (r == c) ? 1e-5f : 0.0f;          // P0 = 1e-5 I
  }
  for (int i = tid; i < NOBS * NS; i += nthr) {
    int r = i / NS, c = i % NS;
    sHt[c][r] = H[i];                             // H^T
  }
  for (int i = tid; i < NS; i += nthr) sx[i] = 0.0f;   // x0 = 0
  // Padded sigma rows (129..143) stay zero so padded sf rows propagate to 0.
  for (int i = tid; i < (NSP - NSIG) * NS; i += nthr) {
    int r = NSIG + i / NS, c = i % NS;
    sig[r][c] = 0.0f;
  }
  __syncthreads();

  float errAcc = 0.0f;

  for (int t = 0; t < TSTEPS; ++t) {
    // --- sW = (n+lam)*(P + 1e-8 I); load y -------------------------------
    for (int i = tid; i < NS * NS; i += nthr) {
      int r = i / NS, c = i % NS;
      sW[r][c] = 63.0f * (sP[r][c] + ((r == c) ? 1e-8f : 0.0f));
    }
    for (int i = tid; i < NOBS; i += nthr)
      sy[i] = Y[(size_t)(traj * TSTEPS + t) * NOBS + i];
    __syncthreads();

    // --- in-place lower Cholesky of sW (right-looking, rank-1 updates) ---
    for (int j = 0; j < NS; ++j) {
      if (tid == 0) sW[j][j] = sqrtf(sW[j][j]);
      __syncthreads();
      float djj = sW[j][j];
      for (int i = j + 1 + tid; i < NS; i += nthr) sW[i][j] /= djj;
      __syncthreads();
      int w2 = NS - 1 - j;
      for (int idx = tid; idx < w2 * w2; idx += nthr) {
        int ii = j + 1 + idx / w2;
        int kk = j + 1 + idx % w2;
        if (kk <= ii) sW[ii][kk] -= sW[ii][j] * sW[kk][j];
      }
      __syncthreads();
    }

    // --- sigma points: sig[0]=x, sig[1+i]=x+L[:,i], sig[1+n+i]=x-L[:,i] --
    for (int i = tid; i < NSIG * NS; i += nthr) {
      int k = i / NS, j = i % NS;
      float base = sx[j];
      if (k == 0) {
        sig[0][j] = base;
      } else if (k <= NS) {
        int r = k - 1;
        sig[k][j] = base + ((j >= r) ? sW[j][r] : 0.0f);
      } else {
        int r = k - 1 - NS;
        sig[k][j] = base - ((j >= r) ? sW[j][r] : 0.0f);
      }
    }
    __syncthreads();

    // --- propagate: sfm = sig @ F^T, then sfm = sig + DT*tanh(sfm) -------
    gemm_ab(&sig[0][0], NS + 1, &sFt[0][0], NS + 1, &sfm[0][0], NS + 1,
            NSP, NS, NS);
    __syncthreads();
    for (int i = tid; i < NSP * NS; i += nthr) {
      int k = i / NS, j = i % NS;
      sfm[k][j] = sig[k][j] + DT_ * tanhf(sfm[k][j]);
    }
    __syncthreads();

    // --- x_pred = Wm @ sf ------------------------------------------------
    if (tid < NS) {
      float acc2 = 0.0f;
      for (int k = 1; k < NSIG; ++k) acc2 += sfm[k][tid];
      sxp[tid] = (-1.0f / 63.0f) * sfm[0][tid] + (0.5f / 63.0f) * acc2;
    }
    // --- zs = sf @ H^T ---------------------------------------------------
    gemm_ab(&sfm[0][0], NS + 1, &sHt[0][0], NOBS + 1, &szs[0][0], NOBS + 1,
            NSP, NOBS, NS);
    __syncthreads();
    // --- z_pred = Wm @ zs ------------------------------------------------
    if (tid < NOBS) {
      float acc2 = 0.0f;
      for (int k = 1; k < NSIG; ++k) acc2 += szs[k][tid];
      szp[tid] = (-1.0f / 63.0f) * szs[0][tid] + (0.5f / 63.0f) * acc2;
    }
    __syncthreads();
    // --- dxs, dzs in place ----------------------------------------------
    for (int i = tid; i < NSP * NS; i += nthr) {
      int k = i / NS, j = i % NS;
      sfm[k][j] -= sxp[j];
    }
    for (int i = tid; i < NSP * NOBS; i += nthr) {
      int k = i / NOBS, j = i % NOBS;
      szs[k][j] -= szp[j];
    }
    __syncthreads();

    // --- weighted Gram matrices (Wc weighting fused into A fragment) -----
    gemm_atb_w(&sfm[0][0], NS + 1, &sfm[0][0], NS + 1, &sW[0][0], NS + 1,
               NS, NS, NSP);                                        // P_pred
    gemm_atb_w(&szs[0][0], NOBS + 1, &szs[0][0], NOBS + 1, &sPz[0][0],
               NOBS + 1, NOBS, NOBS, NSP);                          // Pz
    gemm_atb_w(&sfm[0][0], NS + 1, &szs[0][0], NOBS + 1, &sPxz[0][0],
               NOBS + 1, NS, NOBS, NSP);                            // Pxz
    __syncthreads();
    // add process/measurement noise; stage Pz copy for its Cholesky
    for (int i = tid; i < NS * NS; i += nthr) {
      int r = i / NS, c = i % NS;
      sW[r][c] += Q[i];
    }
    for (int i = tid; i < NOBS * NOBS; i += nthr) {
      int r = i / NOBS, c = i % NOBS;
      float pz = sPz[r][c] + R[i];
      sPz[r][c] = pz;
      sLz[r][c] = pz;
    }
    __syncthreads();

    // --- Cholesky of Pz (32x32, SPD) ------------------------------------
    for (int j = 0; j < NOBS; ++j) {
      if (tid == 0) sLz[j][j] = sqrtf(sLz[j][j]);
      __syncthreads();
      float djj = sLz[j][j];
      for (int i = j + 1 + tid; i < NOBS; i += nthr) sLz[i][j] /= djj;
      __syncthreads();
      int w2 = NOBS - 1 - j;
      for (int idx = tid; idx < w2 * w2; idx += nthr) {
        int ii = j + 1 + idx / w2;
        int kk = j + 1 + idx % w2;
        if (kk <= ii) sLz[ii][kk] -= sLz[ii][j] * sLz[kk][j];
      }
      __syncthreads();
    }

    // --- K = Pxz * Pz^{-1}: two triangular solves per K row --------------
    if (tid < NS) {
      float yv[NOBS];
      for (int i2 = 0; i2 < NOBS; ++i2) {
        float s = sPxz[tid][i2];
        for (int k2 = 0; k2 < i2; ++k2) s -= sLz[i2][k2] * yv[k2];
        yv[i2] = s / sLz[i2][i2];
      }
      for (int i2 = NOBS - 1; i2 >= 0; --i2) {
        float s = yv[i2];
        for (int k2 = i2 + 1; k2 < NOBS; ++k2) s -= sLz[k2][i2] * yv[k2];
        yv[i2] = s / sLz[i2][i2];
      }
      for (int i2 = 0; i2 < NOBS; ++i2) sK[tid][i2] = yv[i2];
    }
    __syncthreads();

    // --- T1 = K @ Pz (reuse zs rows 0..63); T2 = T1 @ K^T (reuse sig) ----
    gemm_ab(&sK[0][0], NOBS + 1, &sPz[0][0], NOBS + 1, &szs[0][0], NOBS + 1,
            NS, NOBS, NOBS);
    __syncthreads();
    gemm_abt(&szs[0][0], NOBS + 1, &sK[0][0], NOBS + 1, &sig[0][0], NS + 1,
             NS, NS, NOBS);
    __syncthreads();

    // --- innovation, state update, MSE accumulation ----------------------
    if (tid < NOBS) sinn[tid] = sy[tid] - szp[tid];
    __syncthreads();
    if (tid < NS) {
      float s = sxp[tid];
      for (int c2 = 0; c2 < NOBS; ++c2) s += sK[tid][c2] * sinn[c2];
      sx[tid] = s;
      out_x[(size_t)(traj * TSTEPS + t) * NS + tid] = s;
      float d = s - X[(size_t)(traj * TSTEPS + t) * NS + tid];
      errAcc += d * d;
    }
    // --- P_new = 0.5*((Pp - T2) + (Pp - T2)^T); write to sP + HBM --------
    for (int i = tid; i < NS * NS; i += nthr) {
      int r = i / NS, c = i % NS;
      float v1 = sW[r][c] - sig[r][c];
      float v2 = sW[c][r] - sig[c][r];
      float pn = 0.5f * (v1 + v2);
      sP[r][c] = pn;
      out_P[((size_t)(traj * TSTEPS + t) * NS + r) * NS + c] = pn;
    }
    __syncthreads();
  }

  // --- per-trajectory squared-error sum -> workspace ----------------------
  if (tid < NS) serr[tid] = errAcc;
  __syncthreads();
  if (tid == 0) {
    float s = 0.0f;
    for (int i = 0; i < NS; ++i) s += serr[i];
    ws_err[traj] = s;
  }
}

// ---------------------------------------------------------------------------
extern "C" __global__ void ukf_mse_kernel(const float* __restrict__ ws_err,
                                          float* __restrict__ tail) {
  __shared__ float sh[NTRAJ];
  int tid = threadIdx.x;
  if (tid < NTRAJ) sh[tid] = ws_err[tid];
  __syncthreads();
  if (tid == 0) {
    float s = 0.0f;
    for (int i = 0; i < NTRAJ; ++i) s += sh[i];
    float mse = s / (float)(NTRAJ * TSTEPS * NS);
    tail[0] = mse;
    tail[1] = 10.0f * log10f(mse);
  }
}

// ---------------------------------------------------------------------------
extern "C" void kernel_launch(void* const* d_in, const int* in_sizes, int n_in,
                              void* d_out, int out_size, void* d_ws,
                              size_t ws_size, hipStream_t stream) {
  (void)in_sizes; (void)n_in; (void)out_size; (void)ws_size;
  const float* X = (const float*)d_in[0];
  const float* Y = (const float*)d_in[1];
  const float* F = (const float*)d_in[2];
  const float* H = (const float*)d_in[3];
  const float* Q = (const float*)d_in[4];
  const float* R = (const float*)d_in[5];

  float* out   = (float*)d_out;
  float* out_x = out;
  float* out_P = out + (size_t)NTRAJ * TSTEPS * NS;
  float* tail  = out_P + (size_t)NTRAJ * TSTEPS * NS * NS;
  float* wse   = (float*)d_ws;

  hipLaunchKernelGGL(ukf_traj_kernel, dim3(NTRAJ), dim3(256), 0, stream,
                     X, Y, F, H, Q, R, out_x, out_P, wse);
  hipLaunchKernelGGL(ukf_mse_kernel, dim3(1), dim3(NTRAJ), 0, stream,
                     wse, tail);
}